// Quanvolution_2860448219693
// MI455X (gfx1250) — compile-verified
//
#include <hip/hip_runtime.h>
#include <math.h>

typedef unsigned int u32;
typedef unsigned long long u64;
typedef __uint128_t u128;

typedef __attribute__((ext_vector_type(2))) float v2f;
typedef __attribute__((ext_vector_type(8))) float v8f;

#define NSITES (32 * 63 * 63) /* 127008, divisible by 16 */

// ---------------------------------------------------------------------------
// numpy default_rng(0) replica: SeedSequence + PCG64 (+ buffered next32)
// ---------------------------------------------------------------------------
struct NpPcg {
  u128 state, inc;
  u32 has32, buf;
};

__device__ inline u128 pcgMult() {
  return (((u128)2549297995355413924ULL) << 64) | (u128)4865540595714422341ULL;
}
__device__ inline void pcgStep(NpPcg& r) { r.state = r.state * pcgMult() + r.inc; }
__device__ inline u64 pcgNext64(NpPcg& r) {
  pcgStep(r);
  u64 hi = (u64)(r.state >> 64), lo = (u64)r.state;
  u64 x = hi ^ lo;
  u32 rot = (u32)(r.state >> 122) & 63u;
  return (x >> rot) | (x << ((64u - rot) & 63u));
}
__device__ inline u32 pcgNext32(NpPcg& r) {
  if (r.has32) { r.has32 = 0; return r.buf; }
  u64 n = pcgNext64(r);
  r.has32 = 1; r.buf = (u32)(n >> 32);
  return (u32)n;
}
__device__ inline double pcgDouble(NpPcg& r) {
  return (double)(pcgNext64(r) >> 11) * (1.0 / 9007199254740992.0);
}
// Generator.integers(high): Lemire 32-bit path, rngIncl = high-1
__device__ inline u32 lemire32(NpPcg& r, u32 rngIncl) {
  u32 rng_excl = rngIncl + 1u;
  u64 m = (u64)pcgNext32(r) * (u64)rng_excl;
  u32 leftover = (u32)m;
  if (leftover < rng_excl) {
    u32 threshold = (0xFFFFFFFFu - rngIncl) % rng_excl;
    while (leftover < threshold) {
      m = (u64)pcgNext32(r) * (u64)rng_excl;
      leftover = (u32)m;
    }
  }
  return (u32)(m >> 32);
}
// random_interval (used by Generator.shuffle): masked rejection, 32-bit path
__device__ inline u32 randInterval(NpPcg& r, u32 mx) {
  if (!mx) return 0;
  u32 mask = mx;
  mask |= mask >> 1; mask |= mask >> 2; mask |= mask >> 4;
  mask |= mask >> 8; mask |= mask >> 16;
  u32 v;
  do { v = pcgNext32(r) & mask; } while (v > mx);
  return v;
}

__device__ inline float2 cmul(float2 a, float2 b) {
  return make_float2(a.x * b.x - a.y * b.y, a.x * b.y + a.y * b.x);
}
__device__ inline float2 cadd(float2 a, float2 b) {
  return make_float2(a.x + b.x, a.y + b.y);
}

// ---------------------------------------------------------------------------
// Setup: regenerate gate structure, build U (16x16 complex), emit
// M[w][m][k] = Re(U^H D_w U) (4 real symmetric 16x16 matrices) into ws.
// ---------------------------------------------------------------------------
__global__ void quanv_setup(const float* __restrict__ params, float* __restrict__ Mout) {
  __shared__ float2 U[16][16]; // U[j][m] = <j|U|m>
  if (threadIdx.x == 0) {
    // SeedSequence(0): entropy = [0], pool_size = 4
    u32 pool[4];
    u32 hc = 0x43b0d7e5u; // INIT_A
    for (int i = 0; i < 4; ++i) {
      u32 v = 0u; // entropy word (0) for i==0, zeros after
      v ^= hc; hc *= 0x931e8875u; v *= hc; v ^= v >> 16; // hashmix
      pool[i] = v;
    }
    for (int s = 0; s < 4; ++s)
      for (int d = 0; d < 4; ++d)
        if (s != d) {
          u32 v = pool[s];
          v ^= hc; hc *= 0x931e8875u; v *= hc; v ^= v >> 16; // hashmix(pool[s])
          u32 res = pool[d] * 0xca01f9ddu - v * 0x4973f715u; // mix
          res ^= res >> 16;
          pool[d] = res;
        }
    // generate_state(4, uint64) -> 8 uint32 words
    u32 st[8];
    u32 hb = 0x8b51f9ddu; // INIT_B
    for (int i = 0; i < 8; ++i) {
      u32 v = pool[i & 3];
      v ^= hb; hb *= 0x58f38dedu; v *= hb; v ^= v >> 16;
      st[i] = v;
    }
    u64 s64[4];
    for (int i = 0; i < 4; ++i) s64[i] = (u64)st[2 * i] | ((u64)st[2 * i + 1] << 32);
    NpPcg r; r.has32 = 0; r.buf = 0;
    u128 initstate = (((u128)s64[0]) << 64) | (u128)s64[1]; // seed[0]=hi, seed[1]=lo
    u128 initseq   = (((u128)s64[2]) << 64) | (u128)s64[3];
    r.inc = (initseq << 1) | (u128)1;
    r.state = 0; pcgStep(r); r.state += initstate; pcgStep(r);

    for (int j = 0; j < 16; ++j)
      for (int m = 0; m < 16; ++m)
        U[j][m] = make_float2(j == m ? 1.f : 0.f, 0.f);

    for (int l = 0; l < 4; ++l) {
      int i = 0;
      while (i < 4) {
        if (pcgDouble(r) > 0.3) {
          int axis = (int)lemire32(r, 2u); // integers(3)
          int wire = (int)lemire32(r, 3u); // integers(4)
          float th = params[l * 4 + i];
          float ch = cosf(0.5f * th), sh = sinf(0.5f * th);
          float2 g00, g01, g10, g11;
          if (axis == 0) {        // RX
            g00 = make_float2(ch, 0.f);  g01 = make_float2(0.f, -sh);
            g10 = make_float2(0.f, -sh); g11 = make_float2(ch, 0.f);
          } else if (axis == 1) { // RY
            g00 = make_float2(ch, 0.f);  g01 = make_float2(-sh, 0.f);
            g10 = make_float2(sh, 0.f);  g11 = make_float2(ch, 0.f);
          } else {                // RZ
            g00 = make_float2(ch, -sh);  g01 = make_float2(0.f, 0.f);
            g10 = make_float2(0.f, 0.f); g11 = make_float2(ch, sh);
          }
          int mb = 1 << (3 - wire);
          for (int m = 0; m < 16; ++m)
            for (int j0 = 0; j0 < 16; ++j0)
              if (!(j0 & mb)) {
                int j1 = j0 | mb;
                float2 a = U[j0][m], bb = U[j1][m];
                U[j0][m] = cadd(cmul(g00, a), cmul(g01, bb));
                U[j1][m] = cadd(cmul(g10, a), cmul(g11, bb));
              }
          ++i;
        } else {
          int perm[4] = {0, 1, 2, 3};
          for (int ii = 3; ii >= 1; --ii) { // Fisher-Yates (Generator.shuffle)
            int j = (int)randInterval(r, (u32)ii);
            int t = perm[ii]; perm[ii] = perm[j]; perm[j] = t;
          }
          int cb = 1 << (3 - perm[0]), tb = 1 << (3 - perm[1]);
          for (int m = 0; m < 16; ++m)
            for (int j0 = 0; j0 < 16; ++j0)
              if ((j0 & cb) && !(j0 & tb)) {
                int j1 = j0 | tb;
                float2 t2 = U[j0][m]; U[j0][m] = U[j1][m]; U[j1][m] = t2;
              }
        }
      }
    }
  }
  __syncthreads();
  for (int rowi = threadIdx.x; rowi < 64; rowi += blockDim.x) {
    int w = rowi >> 4, m = rowi & 15;
    int shift = 3 - w;
    for (int k = 0; k < 16; ++k) {
      float acc = 0.f;
      for (int j = 0; j < 16; ++j) {
        float sgn = ((j >> shift) & 1) ? -1.f : 1.f;
        acc += sgn * (U[j][m].x * U[j][k].x + U[j][m].y * U[j][k].y);
      }
      Mout[rowi * 16 + k] = acc;
    }
  }
}

// ---------------------------------------------------------------------------
// Main: one wave = 16 sites. T_w = M_w * S16 via V_WMMA_F32_16X16X4_F32
// (4 K-chunks), then y_w = column-dot(T_w, S), max-pool 12 -> 4, store.
// Statevector kept as 16 named scalars so all lane-half selection is done on
// VALUES (v_cndmask), never on addresses -> no LDS/scratch demotion.
// ---------------------------------------------------------------------------
__global__ __launch_bounds__(256) void quanv_main(const float* __restrict__ x,
                                                  const float* __restrict__ M,
                                                  float* __restrict__ out) {
  const int lane = threadIdx.x & 31;
  const int waveId = blockIdx.x * (blockDim.x >> 5) + (threadIdx.x >> 5);
  const int row = lane & 15;
  const bool hif = lane >= 16;

  // A operands: ISA 16x4 f32 A layout: lanes<16 -> K={4kc,4kc+1}, lanes>=16 -> K={4kc+2,4kc+3}
  v2f A[4][4];
#pragma unroll
  for (int w = 0; w < 4; ++w)
#pragma unroll
    for (int kc = 0; kc < 4; ++kc) {
      const float* p = M + ((w * 16 + row) * 16 + 4 * kc + (hif ? 2 : 0));
      A[w][kc].x = p[0];
      A[w][kc].y = p[1];
    }

  int site = waveId * 16 + row;
  const bool active = site < NSITES; // wave-uniform (NSITES % 16 == 0)
  int s = active ? site : 0;
  const int b = s / 3969;
  const int r2 = s - b * 3969;
  const int ho = r2 / 63;
  const int wo = r2 - ho * 63;

  float o0 = -3.0e38f, o1 = -3.0e38f, o2 = -3.0e38f, o3 = -3.0e38f;
  const float HALF_PI = 1.57079632679489662f;

#pragma unroll 1
  for (int c = 0; c < 3; ++c) {
    const float* xp = x + (((b * 3 + c) * 64 + ho) * 64 + wo);
    float a0 = HALF_PI * xp[0];
    float a1 = HALF_PI * xp[1];
    float a2 = HALF_PI * xp[64];
    float a3 = HALF_PI * xp[65];
    float c0 = __cosf(a0), s0 = __sinf(a0);
    float c1 = __cosf(a1), s1 = __sinf(a1);
    float c2 = __cosf(a2), s2 = __sinf(a2);
    float c3 = __cosf(a3), s3 = __sinf(a3);
    float p0 = c0 * c1, p1 = c0 * s1, p2 = s0 * c1, p3 = s0 * s1;
    float q0 = c2 * c3, q1 = c2 * s3, q2 = s2 * c3, q3 = s2 * s3;
    // s0[k], k = b0*8 + b1*4 + b2*2 + b3, as named scalars (stay in VGPRs)
    float S0 = p0 * q0,  S1 = p0 * q1,  S2 = p0 * q2,  S3 = p0 * q3;
    float S4 = p1 * q0,  S5 = p1 * q1,  S6 = p1 * q2,  S7 = p1 * q3;
    float S8 = p2 * q0,  S9 = p2 * q1,  S10 = p2 * q2, S11 = p2 * q3;
    float S12 = p3 * q0, S13 = p3 * q1, S14 = p3 * q2, S15 = p3 * q3;

    // B operands (4x16 f32 B layout mirrors A): value selects only
    v2f B0, B1, B2, B3;
    B0.x = hif ? S2 : S0;   B0.y = hif ? S3 : S1;
    B1.x = hif ? S6 : S4;   B1.y = hif ? S7 : S5;
    B2.x = hif ? S10 : S8;  B2.y = hif ? S11 : S9;
    B3.x = hif ? S14 : S12; B3.y = hif ? S15 : S13;
    v2f Bop[4] = {B0, B1, B2, B3};

    // Elements of S needed for the column dot (M rows v + (hif?8:0))
    float E0 = hif ? S8 : S0,   E1 = hif ? S9 : S1;
    float E2 = hif ? S10 : S2,  E3 = hif ? S11 : S3;
    float E4 = hif ? S12 : S4,  E5 = hif ? S13 : S5;
    float E6 = hif ? S14 : S6,  E7 = hif ? S15 : S7;

#pragma unroll
    for (int w = 0; w < 4; ++w) {
      v8f D = {0.f, 0.f, 0.f, 0.f, 0.f, 0.f, 0.f, 0.f};
#pragma unroll
      for (int kc = 0; kc < 4; ++kc)
        D = __builtin_amdgcn_wmma_f32_16x16x4_f32(false, A[w][kc], false, Bop[kc],
                                                  (short)0, D, false, false);
      // y_w[col] = sum_m T_w[m][col] * S[m]
      float partial = D[0] * E0 + D[1] * E1 + D[2] * E2 + D[3] * E3 +
                      D[4] * E4 + D[5] * E5 + D[6] * E6 + D[7] * E7;
      partial += __shfl_xor(partial, 16, 32);
      // adaptive max-pool 12 -> 4: bin = (c*4 + w) / 3, via value selects
      const int idx = c * 4 + w; // w is compile-time, c runtime
      o0 = (idx <= 2)             ? fmaxf(o0, partial) : o0;
      o1 = (idx >= 3 && idx <= 5) ? fmaxf(o1, partial) : o1;
      o2 = (idx >= 6 && idx <= 8) ? fmaxf(o2, partial) : o2;
      o3 = (idx >= 9)             ? fmaxf(o3, partial) : o3;
    }
  }

  if (active && !hif) {
    float* po = out + b * (4 * 3969) + ho * 63 + wo;
    po[0 * 3969] = o0;
    po[1 * 3969] = o1;
    po[2 * 3969] = o2;
    po[3 * 3969] = o3;
  }
}

// ---------------------------------------------------------------------------
extern "C" void kernel_launch(void* const* d_in, const int* in_sizes, int n_in,
                              void* d_out, int out_size, void* d_ws, size_t ws_size,
                              hipStream_t stream) {
  (void)in_sizes; (void)n_in; (void)out_size; (void)ws_size;
  const float* x = (const float*)d_in[0];
  const float* params = (const float*)d_in[1];
  float* out = (float*)d_out;
  float* Mws = (float*)d_ws; // 4*16*16 floats = 4KB

  quanv_setup<<<1, 32, 0, stream>>>(params, Mws);

  const int nWaves = (NSITES + 15) / 16;       // 7938
  const int nBlocks = (nWaves + 7) / 8;        // 993 (256 thr = 8 waves)
  quanv_main<<<nBlocks, 256, 0, stream>>>(x, Mws, out);
}